// GNNExtrapolation_55198919688664
// MI455X (gfx1250) — compile-verified
//
#include <hip/hip_runtime.h>
#include <hip/hip_bf16.h>

typedef __attribute__((ext_vector_type(2))) float v2f;
typedef __attribute__((ext_vector_type(8))) float v8f;

#define T_IN   12
#define T_OUTX 24
#define C_DIM  16
#define K_NB   17
#define H_HEADS 4
#define KTOT   48   // T_IN * H_HEADS
#define J_OUT  12   // T_OUTX - T_IN

// ---------------------------------------------------------------------------
// Kernel 1: sigma = max(nearest_dists). dists are uniform[0,1) -> nonnegative,
// so float bits compare monotonically as unsigned ints -> atomicMax on u32.
// ---------------------------------------------------------------------------
__global__ void sigma_max_kernel(const float* __restrict__ d, int n,
                                 unsigned int* __restrict__ out_bits) {
    __shared__ float red[256];
    float m = 0.0f;
    for (int i = blockIdx.x * blockDim.x + threadIdx.x; i < n;
         i += gridDim.x * blockDim.x)
        m = fmaxf(m, d[i]);
    red[threadIdx.x] = m;
    __syncthreads();
    for (int s = 128; s > 0; s >>= 1) {
        if ((int)threadIdx.x < s)
            red[threadIdx.x] = fmaxf(red[threadIdx.x], red[threadIdx.x + s]);
        __syncthreads();
    }
    if (threadIdx.x == 0)
        atomicMax(out_bits, __float_as_uint(red[0]));
}

// ---------------------------------------------------------------------------
// Kernel 2: out[b, 0:12, :, :] = x   (vectorized float4 copy, coalesced)
// ---------------------------------------------------------------------------
__global__ void copy_x_kernel(const float4* __restrict__ x,
                              float4* __restrict__ out,
                              int per_b4,        // 12*N*16/4
                              int out_b_stride4, // 24*N*16/4
                              int total4) {
    int i = blockIdx.x * blockDim.x + threadIdx.x;
    if (i < total4) {
        int b = i / per_b4;
        int r = i - b * per_b4;
        out[(size_t)b * out_b_stride4 + r] = x[i];
    }
}

// ---------------------------------------------------------------------------
// Kernel 3: fused gather-aggregate + WMMA shrink matmul + SELU.
// One wave32 per (b, n) tile. Grid = B*N.
// ---------------------------------------------------------------------------
__global__ __launch_bounds__(32) void gnn_main_kernel(
    const float* __restrict__ x,        // (B, 12, N, 16)
    const int*   __restrict__ nbr,      // (N, 17)
    const float* __restrict__ nd,       // (N, 17)
    const float* __restrict__ sw,       // (12, 48)
    const float* __restrict__ sb,       // (12,)
    const float* __restrict__ sigma_p,  // scalar in ws
    float* __restrict__ out,            // (B, 24, N, 16)
    int N) {
    __shared__ float a_lds[KTOT * C_DIM];       // A tile, layout [K][M=c]
    __shared__ float w_lds[K_NB][H_HEADS];
    __shared__ int   idx_lds[K_NB];

    const int tile = blockIdx.x;
    const int n    = tile % N;
    const int b    = tile / N;
    const int lane = threadIdx.x;

    const float sigma   = *sigma_p;
    const float inv_s2  = 1.0f / (sigma * sigma);

    // --- phase 1: per-neighbor weights + indices -------------------------
    if (lane < K_NB) {
        const int   node    = nbr[n * K_NB + lane];
        const float d       = nd[n * K_NB + lane];
        const bool  invalid = (node == -1);
        idx_lds[lane] = invalid ? 0 : node;
        const float d2s = d * d * inv_s2;
#pragma unroll
        for (int h = 0; h < H_HEADS; ++h) {
            const float lam = (float)(h + 1) * (1.0f / (float)H_HEADS);
            float wv = expf(-d2s * lam);
            if (invalid || wv < 1e-8f) wv = 0.0f;
            w_lds[lane][h] = wv;
        }
    }
    __syncthreads();

    // --- phase 2: gather + weighted aggregation --------------------------
    // lane -> (c = lane&15, t-half = lane>>4); 6 iters cover t = 0..11.
    const int c     = lane & 15;
    const int thalf = lane >> 4;

    float acc[6][H_HEADS];
#pragma unroll
    for (int it = 0; it < 6; ++it)
#pragma unroll
        for (int h = 0; h < H_HEADS; ++h) acc[it][h] = 0.0f;

    const float* xb = x + (size_t)b * T_IN * N * C_DIM;

    for (int k = 0; k < K_NB; ++k) {
        const int   node = idx_lds[k];
        const float w0 = w_lds[k][0], w1 = w_lds[k][1];
        const float w2 = w_lds[k][2], w3 = w_lds[k][3];
#pragma unroll
        for (int it = 0; it < 6; ++it) {
            const int t = it * 2 + thalf;
            const float v = xb[((size_t)t * N + node) * C_DIM + c];
            acc[it][0] = fmaf(v, w0, acc[it][0]);
            acc[it][1] = fmaf(v, w1, acc[it][1]);
            acc[it][2] = fmaf(v, w2, acc[it][2]);
            acc[it][3] = fmaf(v, w3, acc[it][3]);
        }
    }

    // --- phase 3: stage A tile in LDS: A[M=c][K = t*4+h], stored [K][c] --
#pragma unroll
    for (int it = 0; it < 6; ++it) {
        const int t = it * 2 + thalf;
#pragma unroll
        for (int h = 0; h < H_HEADS; ++h)
            a_lds[(t * H_HEADS + h) * C_DIM + c] = acc[it][h];
    }
    __syncthreads();

    // --- phase 4: 16x48 @ 48x16 via 12 chained V_WMMA_F32_16X16X4_F32 ----
    // A frag: lanes 0-15 -> row M=lane, K pair (4j,4j+1); lanes 16-31 ->
    // row M=lane-16, K pair (4j+2,4j+3). B frag mirrors with N columns.
    // All fragments preloaded (branch-free, masked-multiply padding), then
    // the 12 WMMAs run as a pure back-to-back accumulate chain.
    const int row  = lane & 15;
    const int koff = (lane >> 4) * 2;

    const int   safe_col = (row < J_OUT) ? row : 0;
    const float colmask  = (row < J_OUT) ? 1.0f : 0.0f;
    const float* swc     = sw + safe_col * KTOT;

    v2f afrag[12], bfrag[12];
#pragma unroll
    for (int j = 0; j < 12; ++j) {
        const int k0 = j * 4 + koff;
        afrag[j].x = a_lds[(k0 + 0) * C_DIM + row];
        afrag[j].y = a_lds[(k0 + 1) * C_DIM + row];
        bfrag[j].x = colmask * swc[k0 + 0];
        bfrag[j].y = colmask * swc[k0 + 1];
    }

    v8f d_acc = {};
#pragma unroll
    for (int j = 0; j < 12; ++j) {
        d_acc = __builtin_amdgcn_wmma_f32_16x16x4_f32(
            /*neg_a=*/false, afrag[j], /*neg_b=*/false, bfrag[j],
            /*c_mod=*/(short)0, d_acc, /*reuse_a=*/false, /*reuse_b=*/false);
    }

    // --- phase 5: bias + SELU + store ------------------------------------
    // D layout: lane L -> column N = L&15; VGPR r -> row M = r (+8 if L>=16)
    const int col = lane & 15;
    if (col < J_OUT) {
        const float bias  = sb[col];
        const int   mbase = (lane < 16) ? 0 : 8;
        float* outp = out + (((size_t)b * T_OUTX + T_IN + col) * N + n) * C_DIM;
        const float kScale = 1.0507009873554804934193349852946f;
        const float kAlpha = 1.6732632423543772848170429916717f;
#pragma unroll
        for (int r = 0; r < 8; ++r) {
            const float y = d_acc[r] + bias;
            const float s = (y > 0.0f) ? kScale * y
                                       : kScale * kAlpha * (expf(y) - 1.0f);
            outp[mbase + r] = s;
        }
    }
}

// ---------------------------------------------------------------------------
extern "C" void kernel_launch(void* const* d_in, const int* in_sizes, int n_in,
                              void* d_out, int out_size, void* d_ws, size_t ws_size,
                              hipStream_t stream) {
    const float* x   = (const float*)d_in[0];
    const int*   nbr = (const int*)d_in[1];
    const float* nd  = (const float*)d_in[2];
    const float* sw  = (const float*)d_in[3];
    const float* sb  = (const float*)d_in[4];
    float*       out = (float*)d_out;

    const int NK = in_sizes[1];          // N * 17
    const int N  = NK / K_NB;            // 10000
    const int B  = in_sizes[0] / (T_IN * N * C_DIM);  // 2

    unsigned int* sig_bits = (unsigned int*)d_ws;

    // sigma = max(nearest_dists)
    hipMemsetAsync(d_ws, 0, sizeof(unsigned int), stream);
    {
        const int cnt    = in_sizes[2];  // N*17
        int       blocks = (cnt + 255) / 256;
        if (blocks > 1024) blocks = 1024;
        sigma_max_kernel<<<blocks, 256, 0, stream>>>(nd, cnt, sig_bits);
    }

    // out[:, :12] = x
    {
        const int per_b4        = T_IN * N * C_DIM / 4;
        const int out_b_stride4 = T_OUTX * N * C_DIM / 4;
        const int total4        = B * per_b4;
        const int blocks        = (total4 + 255) / 256;
        copy_x_kernel<<<blocks, 256, 0, stream>>>(
            (const float4*)x, (float4*)out, per_b4, out_b_stride4, total4);
    }

    // fused gather/aggregate + WMMA shrink + SELU -> out[:, 12:]
    gnn_main_kernel<<<B * N, 32, 0, stream>>>(
        x, nbr, nd, sw, sb, (const float*)d_ws, out, N);
}